// PcPosEnc_41154376630650
// MI455X (gfx1250) — compile-verified
//
#include <hip/hip_runtime.h>
#include <hip/hip_bf16.h>

#define DM    256
#define KNN   16
#define NPTS  2048
#define NB    8
#define NQ    (NB * NPTS)          // 16384 query points
#define W2PITCH 260                // 256 + 4 pad floats -> conflict-free B-operand ds_load_b64
#define W2T_OFF  0
#define W1B1_OFF (W2PITCH * 256)   // float index
#define B2_OFF   (W1B1_OFF + 4 * 256)
#define SMEM_FLOATS (B2_OFF + 256) // 67840 floats = 271360 bytes < 320KB WGP LDS

typedef __attribute__((ext_vector_type(2))) float v2f;
typedef __attribute__((ext_vector_type(8))) float v8f;

// ---------------------------------------------------------------------------
// Kernel 1: exact KNN (stable top-16 by distance, ascending) + rel = p - knn
// One block = 256 queries of one batch; batch's points staged in LDS as
// float4 {x, y, z, |p|^2}. dist = sq_n + sq_m - 2*dot (same formula as ref,
// self-distance exactly 0).
// ---------------------------------------------------------------------------
__global__ __launch_bounds__(256)
void knn_rel_kernel(const float* __restrict__ pc,
                    float* __restrict__ relbuf, long relStride)
{
    __shared__ float4 pts[NPTS];
    const int b     = blockIdx.x >> 3;
    const int chunk = blockIdx.x & 7;
    const int tid   = threadIdx.x;

    const float* pcb = pc + (size_t)b * NPTS * 3;
    for (int i = tid; i < NPTS; i += 256) {
        float x = pcb[i * 3 + 0];
        float y = pcb[i * 3 + 1];
        float z = pcb[i * 3 + 2];
        float sq = fmaf(z, z, fmaf(y, y, x * x));
        pts[i] = make_float4(x, y, z, sq);
    }
    __syncthreads();

    const int n = chunk * 256 + tid;
    const float4 own = pts[n];

    float bd[KNN];
    int   bi[KNN];
#pragma unroll
    for (int j = 0; j < KNN; ++j) { bd[j] = 3.0e38f; bi[j] = 0; }

    for (int m = 0; m < NPTS; ++m) {
        float4 p = pts[m];
        float dot = fmaf(own.z, p.z, fmaf(own.y, p.y, own.x * p.x));
        float d = own.w + p.w - 2.0f * dot;
        if (d < bd[KNN - 1]) {
            // branch-free unrolled stable insertion (registers only)
#pragma unroll
            for (int j = KNN - 1; j >= 0; --j) {
                bool lt = d < bd[j];
                bool ge_prev = (j == 0) || !(d < bd[j - 1]);
                if (lt) {
                    if (ge_prev) { bd[j] = d; bi[j] = m; }
                    else         { bd[j] = bd[j - 1]; bi[j] = bi[j - 1]; }
                }
            }
        }
    }

    float* rq = relbuf + (size_t)(b * NPTS + n) * relStride;
#pragma unroll
    for (int j = 0; j < KNN; ++j) {
        float4 p = pts[bi[j]];
        rq[j * 3 + 0] = own.x - p.x;
        rq[j * 3 + 1] = own.y - p.y;
        rq[j * 3 + 2] = own.z - p.z;
    }
}

// ---------------------------------------------------------------------------
// Kernel 2: fused MLP. One wave per query point (M-tile of 16 neighbor rows).
// h = relu(rel@W1 + b1) computed in VALU directly in the WMMA A-operand
// layout (lane = M row, VGPR pair = K pair selected by lane half).
// out = h@W2 + b2 via V_WMMA_F32_16X16X4_F32, W2 transposed+padded in LDS,
// bias preloaded into the C accumulator. Two N-tiles interleaved for ILP.
// ---------------------------------------------------------------------------
__global__ __launch_bounds__(256, 1)
void fused_mlp_kernel(const float* __restrict__ relbuf, long relStride,
                      const float* __restrict__ W1, const float* __restrict__ b1,
                      const float* __restrict__ W2, const float* __restrict__ b2,
                      float* __restrict__ out, int numWavesTotal)
{
    extern __shared__ float smem[];
    const int tid = threadIdx.x;

    // Stage W2^T with pitch 260 (banks (4n+k)%64 -> conflict-free b64 reads)
    for (int i = tid; i < DM * DM; i += 256) {
        int k = i >> 8;
        int nn = i & 255;
        smem[W2T_OFF + nn * W2PITCH + k] = W2[i];
    }
    // Pack {W1[0][k], W1[1][k], W1[2][k], b1[k]} as float4; b2 plain.
    for (int k = tid; k < DM; k += 256) {
        smem[W1B1_OFF + 4 * k + 0] = W1[0 * DM + k];
        smem[W1B1_OFF + 4 * k + 1] = W1[1 * DM + k];
        smem[W1B1_OFF + 4 * k + 2] = W1[2 * DM + k];
        smem[W1B1_OFF + 4 * k + 3] = b1[k];
        smem[B2_OFF + k] = b2[k];
    }
    __syncthreads();

    const int lane = tid & 31;
    const int wave = tid >> 5;
    const int half = lane >> 4;    // which 16-lane half
    const int mrow = lane & 15;    // A: M row; B/C: N column
    const int gwave = blockIdx.x * 8 + wave;

    for (int q = gwave; q < NQ; q += numWavesTotal) {
        // rel row for this lane's M row (both halves load the same row)
        const float* rp = relbuf + (size_t)q * relStride + mrow * 3;
        float r0 = rp[0], r1 = rp[1], r2 = rp[2];

        // h in A-matrix layout: chunk c -> K = 4c + 2*half + {0,1}
        float h[128];
#pragma unroll
        for (int c = 0; c < 64; ++c) {
            int k0 = 4 * c + 2 * half;
            float4 wa = *(const float4*)&smem[W1B1_OFF + 4 * k0];
            float4 wb = *(const float4*)&smem[W1B1_OFF + 4 * k0 + 4];
            float ha = fmaf(r2, wa.z, fmaf(r1, wa.y, fmaf(r0, wa.x, wa.w)));
            float hb = fmaf(r2, wb.z, fmaf(r1, wb.y, fmaf(r0, wb.x, wb.w)));
            h[2 * c]     = fmaxf(ha, 0.0f);
            h[2 * c + 1] = fmaxf(hb, 0.0f);
        }

        float* outq = out + (size_t)q * (KNN * DM);

#pragma unroll 1
        for (int tp = 0; tp < 8; ++tp) {
            const int t0 = 2 * tp, t1 = 2 * tp + 1;
            const float bias0 = smem[B2_OFF + t0 * 16 + mrow];
            const float bias1 = smem[B2_OFF + t1 * 16 + mrow];
            v8f acc0, acc1;
#pragma unroll
            for (int v = 0; v < 8; ++v) { acc0[v] = bias0; acc1[v] = bias1; }

            const float* bb0 = &smem[W2T_OFF + (t0 * 16 + mrow) * W2PITCH + 2 * half];
            const float* bb1 = &smem[W2T_OFF + (t1 * 16 + mrow) * W2PITCH + 2 * half];

#pragma unroll
            for (int c = 0; c < 64; ++c) {
                v2f a;
                a[0] = h[2 * c];
                a[1] = h[2 * c + 1];
                v2f bv0 = *(const v2f*)(bb0 + 4 * c);
                v2f bv1 = *(const v2f*)(bb1 + 4 * c);
                acc0 = __builtin_amdgcn_wmma_f32_16x16x4_f32(
                    false, a, false, bv0, (short)0, acc0, false, false);
                acc1 = __builtin_amdgcn_wmma_f32_16x16x4_f32(
                    false, a, false, bv1, (short)0, acc1, false, false);
            }

            // C layout: VGPR v, M = v + 8*half, N = tile*16 + mrow
#pragma unroll
            for (int v = 0; v < 8; ++v) {
                int m = v + 8 * half;
                outq[m * DM + t0 * 16 + mrow] = acc0[v];
                outq[m * DM + t1 * 16 + mrow] = acc1[v];
            }
        }
    }
}

extern "C" void kernel_launch(void* const* d_in, const int* in_sizes, int n_in,
                              void* d_out, int out_size, void* d_ws, size_t ws_size,
                              hipStream_t stream)
{
    const float* pc = (const float*)d_in[0];
    const float* W1 = (const float*)d_in[1];
    const float* b1 = (const float*)d_in[2];
    const float* W2 = (const float*)d_in[3];
    const float* b2 = (const float*)d_in[4];
    float* out = (float*)d_out;

    // rel scratch: prefer d_ws; else stash in each query's own output block
    // (kernel 2 reads its query's rel before overwriting that block -> safe).
    const size_t relBytes = (size_t)NQ * KNN * 3 * sizeof(float);
    float* relbuf;
    long relStride;
    if (ws_size >= relBytes) { relbuf = (float*)d_ws; relStride = KNN * 3; }
    else                     { relbuf = out;          relStride = KNN * DM; }

    knn_rel_kernel<<<NB * 8, 256, 0, stream>>>(pc, relbuf, relStride);

    const int nBlocks = 256;              // persistent: 8 queries per wave
    const int numWaves = nBlocks * 8;
    const size_t smemBytes = SMEM_FLOATS * sizeof(float);  // 271360 B
    hipFuncSetAttribute(reinterpret_cast<const void*>(fused_mlp_kernel),
                        hipFuncAttributeMaxDynamicSharedMemorySize,
                        (int)smemBytes);
    fused_mlp_kernel<<<nBlocks, 256, smemBytes, stream>>>(
        relbuf, relStride, W1, b1, W2, b2, out, numWaves);
}